// GCNN_64836826301090
// MI455X (gfx1250) — compile-verified
//
#include <hip/hip_runtime.h>

typedef __attribute__((ext_vector_type(2))) float v2f;
typedef __attribute__((ext_vector_type(8))) float v8f;

#define FDIM 64

// ---------------- degree / normalization precompute ----------------

__global__ __launch_bounds__(256) void k_init_deg(float* __restrict__ deg, int n) {
    int i = blockIdx.x * blockDim.x + threadIdx.x;
    if (i < n) deg[i] = 1.0f;              // self-loop weight
}

__global__ __launch_bounds__(256) void k_deg_accum(const int* __restrict__ dst,
                                                   const float* __restrict__ w,
                                                   float* __restrict__ deg, int e) {
    int i = blockIdx.x * blockDim.x + threadIdx.x;
    if (i < e) atomicAdd(&deg[dst[i]], w[i]);
}

__global__ __launch_bounds__(256) void k_dinv(float* __restrict__ deg_dinv,
                                              float* __restrict__ self_norm, int n) {
    int i = blockIdx.x * blockDim.x + threadIdx.x;
    if (i < n) {
        float d = rsqrtf(deg_dinv[i]);
        deg_dinv[i]  = d;       // becomes dinv in place
        self_norm[i] = d * d;
    }
}

__global__ __launch_bounds__(256) void k_norm(const int* __restrict__ src,
                                              const int* __restrict__ dst,
                                              const float* __restrict__ w,
                                              const float* __restrict__ dinv,
                                              float* __restrict__ norm, int e) {
    int i = blockIdx.x * blockDim.x + threadIdx.x;
    if (i < e) norm[i] = dinv[src[i]] * w[i] * dinv[dst[i]];
}

// ---------------- fp32 GEMM: out[N,64] = H[N,64] @ W[64,64] via V_WMMA_F32_16X16X4_F32 ----

__global__ __launch_bounds__(256) void k_gemm64_wmma(const float* __restrict__ H,
                                                     const float* __restrict__ Wm,
                                                     float* __restrict__ out, int nrows) {
    const int lane  = threadIdx.x & 31;
    const int wave  = threadIdx.x >> 5;
    const int ctile = wave & 3;     // 4 column tiles of 16
    const int rhalf = wave >> 2;    // 2 row tiles per block -> block covers 32 rows
    const int l16   = lane & 15;
    const int hi    = lane >> 4;    // half-wave selects K pair {0,1} or {2,3}

    // Preload B fragments (4x16 tiles of W, K-major) once per wave: 16 steps x 2 VGPRs.
    v2f bfrag[16];
    {
        const int col = ctile * 16 + l16;
        #pragma unroll
        for (int k = 0; k < 16; ++k) {
            const int k0 = 4 * k + (hi ? 2 : 0);
            bfrag[k].x = Wm[(size_t)k0 * FDIM + col];
            bfrag[k].y = Wm[(size_t)(k0 + 1) * FDIM + col];
        }
    }

    const int cc = ctile * 16 + l16;
    const int rowBlocks = (nrows + 31) >> 5;
    for (int rb = blockIdx.x; rb < rowBlocks; rb += gridDim.x) {
        const int row0 = rb * 32 + rhalf * 16;          // wave-uniform
        const bool full = (row0 + 16 <= nrows);         // wave-uniform fast path

        int r  = row0 + l16;                            // A-matrix: lane%16 == M index
        int rc = full ? r : ((r < nrows) ? r : (nrows > 0 ? nrows - 1 : 0));
        const float* hp = H + (size_t)rc * FDIM + (hi ? 2 : 0);

        v8f acc = {};
        #pragma unroll
        for (int k = 0; k < 16; ++k) {
            v2f a = *(const v2f*)(hp + 4 * k);   // 8B-aligned: offsets 16k / 16k+8
            acc = __builtin_amdgcn_wmma_f32_16x16x4_f32(
                      false, a, false, bfrag[k], (short)0, acc, false, false);
        }

        // C/D layout: VGPR i -> lanes 0-15: M=i, lanes 16-31: M=8+i
        if (full) {
            float* op = out + (size_t)(row0 + hi * 8) * FDIM + cc;
            #pragma unroll
            for (int i = 0; i < 8; ++i)
                op[(size_t)i * FDIM] = acc[i];          // imm offsets i*256B -> store clause
        } else {
            #pragma unroll
            for (int i = 0; i < 8; ++i) {
                int rr = row0 + hi * 8 + i;
                if (rr < nrows) out[(size_t)rr * FDIM + cc] = acc[i];
            }
        }
    }
}

// ---------------- epilogue: self-loop + bias, edge scatter, relu ----------------

__global__ __launch_bounds__(256) void k_epi_init(const float* __restrict__ hw,
                                                  const float* __restrict__ self_norm,
                                                  const float* __restrict__ b,
                                                  float* __restrict__ out, int n) {
    long gid = (long)blockIdx.x * blockDim.x + threadIdx.x;
    if (gid < (long)n * FDIM) {
        int i = (int)(gid >> 6);
        out[gid] = hw[gid] * self_norm[i] + b[gid & 63];
    }
}

__global__ __launch_bounds__(256) void k_scatter(const float* __restrict__ hw,
                                                 const float* __restrict__ norm,
                                                 const int* __restrict__ src,
                                                 const int* __restrict__ dst,
                                                 float* __restrict__ out, int e) {
    long gid = (long)blockIdx.x * blockDim.x + threadIdx.x;   // 16 threads / edge
    if (gid >= (long)e * 16) return;
    int  ed = (int)(gid >> 4);
    int  g  = ((int)gid & 15) * 4;
    float nv = norm[ed];
    int   s  = src[ed];
    int   d  = dst[ed];
    const float4 v = *(const float4*)(hw + (size_t)s * FDIM + g);
    float* op = out + (size_t)d * FDIM + g;
    atomicAdd(op + 0, v.x * nv);
    atomicAdd(op + 1, v.y * nv);
    atomicAdd(op + 2, v.z * nv);
    atomicAdd(op + 3, v.w * nv);
}

__global__ __launch_bounds__(256) void k_relu(float* __restrict__ h, long total) {
    long gid = (long)blockIdx.x * blockDim.x + threadIdx.x;
    if (gid < total) h[gid] = fmaxf(h[gid], 0.0f);
}

// ---------------- final 64 -> 1 layer ----------------

__global__ __launch_bounds__(256) void k_matvec64(const float* __restrict__ H,
                                                  const float* __restrict__ w2,
                                                  float* __restrict__ lin, int n) {
    int i = blockIdx.x * blockDim.x + threadIdx.x;
    if (i < n) {
        const float* hp = H + (size_t)i * FDIM;
        float s = 0.0f;
        #pragma unroll
        for (int f = 0; f < FDIM; ++f) s = fmaf(hp[f], w2[f], s);
        lin[i] = s;
    }
}

__global__ __launch_bounds__(256) void k_init2(const float* __restrict__ lin,
                                               const float* __restrict__ self_norm,
                                               const float* __restrict__ b2,
                                               float* __restrict__ out, int n) {
    int i = blockIdx.x * blockDim.x + threadIdx.x;
    if (i < n) out[i] = lin[i] * self_norm[i] + b2[0];
}

__global__ __launch_bounds__(256) void k_scatter2(const float* __restrict__ lin,
                                                  const float* __restrict__ norm,
                                                  const int* __restrict__ src,
                                                  const int* __restrict__ dst,
                                                  float* __restrict__ out, int e) {
    int i = blockIdx.x * blockDim.x + threadIdx.x;
    if (i < e) atomicAdd(&out[dst[i]], lin[src[i]] * norm[i]);
}

// ---------------- driver ----------------

extern "C" void kernel_launch(void* const* d_in, const int* in_sizes, int n_in,
                              void* d_out, int out_size, void* d_ws, size_t ws_size,
                              hipStream_t stream) {
    const float* x    = (const float*)d_in[0];
    const int*   esrc = (const int*)  d_in[1];
    const int*   edst = (const int*)  d_in[2];
    const float* ew   = (const float*)d_in[3];
    const float* W0   = (const float*)d_in[4];
    const float* b0   = (const float*)d_in[5];
    const float* W1   = (const float*)d_in[6];
    const float* b1   = (const float*)d_in[7];
    const float* W2   = (const float*)d_in[8];
    const float* b2   = (const float*)d_in[9];

    const int N = in_sizes[0] / FDIM;
    const int E = in_sizes[1];
    float* out = (float*)d_out;

    // workspace layout (floats): dinv[N] | self_norm[N] | norm[E] | hw[N*64] | ha[N*64]
    float* ws        = (float*)d_ws;
    float* dinv      = ws;
    float* self_norm = ws + (size_t)N;
    float* normw     = ws + 2 * (size_t)N;
    float* hw        = normw + (size_t)E;
    float* ha        = hw + (size_t)N * FDIM;

    const int T = 256;
    const int gN  = (N + T - 1) / T;
    const int gE  = (E + T - 1) / T;
    const long nf = (long)N * FDIM;
    const int gNF = (int)((nf + T - 1) / T);
    const long es = (long)E * 16;
    const int gES = (int)((es + T - 1) / T);
    const int gGemm = (N + 31) / 32;

    // normalization precompute
    k_init_deg <<<gN, T, 0, stream>>>(dinv, N);
    k_deg_accum<<<gE, T, 0, stream>>>(edst, ew, dinv, E);
    k_dinv     <<<gN, T, 0, stream>>>(dinv, self_norm, N);
    k_norm     <<<gE, T, 0, stream>>>(esrc, edst, ew, dinv, normw, E);

    // layer 0: relu(conv(x, W0, b0)) -> ha
    k_gemm64_wmma<<<gGemm, T, 0, stream>>>(x, W0, hw, N);
    k_epi_init   <<<gNF, T, 0, stream>>>(hw, self_norm, b0, ha, N);
    k_scatter    <<<gES, T, 0, stream>>>(hw, normw, esrc, edst, ha, E);
    k_relu       <<<gNF, T, 0, stream>>>(ha, nf);

    // layer 1: relu(conv(ha, W1, b1)) -> ha (in place across kernel boundaries)
    k_gemm64_wmma<<<gGemm, T, 0, stream>>>(ha, W1, hw, N);
    k_epi_init   <<<gNF, T, 0, stream>>>(hw, self_norm, b1, ha, N);
    k_scatter    <<<gES, T, 0, stream>>>(hw, normw, esrc, edst, ha, E);
    k_relu       <<<gNF, T, 0, stream>>>(ha, nf);

    // layer 2: conv(ha, W2, b2) -> out[N]
    float* lin = hw;   // reuse
    k_matvec64<<<gN, T, 0, stream>>>(ha, W2, lin, N);
    k_init2   <<<gN, T, 0, stream>>>(lin, self_norm, b2, out, N);
    k_scatter2<<<gE, T, 0, stream>>>(lin, normw, esrc, edst, out, E);
}